// PlanLayer_41515153883715
// MI455X (gfx1250) — compile-verified
//
#include <hip/hip_runtime.h>
#include <math.h>

#define Bsz   128
#define T1c   128
#define SDc   4
#define NGPc  127
#define HSc   128
#define WSc   128
#define DTc   (5.0f/127.0f)
#define REGc  0.001f
#define KSIc  10000.0f      // 1/KS^2 == 1/KG^2
#define HLF   63.5f         // (WS-1)/2
#define NBAT  4             // batches per workgroup (packed into one WMMA)
#define NTH   512

typedef float v2f __attribute__((ext_vector_type(2)));
typedef float v8f __attribute__((ext_vector_type(8)));

// Qinv 4x4 from the 2x2 qc_inv block: [[c3*q, c2*q],[c2*q, c1*q]]
__device__ __forceinline__ void buildQ(const float* __restrict__ q, float Q[4][4]) {
  const float c3 = 12.0f/(DTc*DTc*DTc);
  const float c2 = -6.0f/(DTc*DTc);
  const float c1 =  4.0f/DTc;
  const float q00=q[0], q01=q[1], q10=q[2], q11=q[3];
  Q[0][0]=c3*q00; Q[0][1]=c3*q01; Q[0][2]=c2*q00; Q[0][3]=c2*q01;
  Q[1][0]=c3*q10; Q[1][1]=c3*q11; Q[1][2]=c2*q10; Q[1][3]=c2*q11;
  Q[2][0]=c2*q00; Q[2][1]=c2*q01; Q[2][2]=c1*q00; Q[2][3]=c1*q01;
  Q[3][0]=c2*q10; Q[3][1]=c2*q11; Q[3][2]=c1*q10; Q[3][3]=c1*q11;
}

// In-place inverse of a 4x4 SPD matrix (row-major, 16 floats in LDS) via Cholesky.
__device__ __forceinline__ void inv4_spd(float* __restrict__ s) {
  const float a00=s[0],  a10=s[4],  a11=s[5],  a20=s[8],  a21=s[9],  a22=s[10];
  const float a30=s[12], a31=s[13], a32=s[14], a33=s[15];
  const float L00 = sqrtf(fmaxf(a00, 1e-30f));            const float i00 = 1.0f/L00;
  const float L10 = a10*i00, L20 = a20*i00, L30 = a30*i00;
  const float L11 = sqrtf(fmaxf(a11 - L10*L10, 1e-30f));  const float i11 = 1.0f/L11;
  const float L21 = (a21 - L20*L10)*i11;
  const float L31 = (a31 - L30*L10)*i11;
  const float L22 = sqrtf(fmaxf(a22 - L20*L20 - L21*L21, 1e-30f)); const float i22 = 1.0f/L22;
  const float L32 = (a32 - L30*L20 - L31*L21)*i22;
  const float L33 = sqrtf(fmaxf(a33 - L30*L30 - L31*L31 - L32*L32, 1e-30f)); const float i33 = 1.0f/L33;
  // Li = L^{-1} (lower triangular)
  const float B00 = i00;
  const float B10 = -i11*(L10*B00);
  const float B11 = i11;
  const float B20 = -i22*(L20*B00 + L21*B10);
  const float B21 = -i22*(L21*B11);
  const float B22 = i22;
  const float B30 = -i33*(L30*B00 + L31*B10 + L32*B20);
  const float B31 = -i33*(L31*B11 + L32*B21);
  const float B32 = -i33*(L32*B22);
  const float B33 = i33;
  // M = Li^T * Li (symmetric)
  const float M00 = B00*B00 + B10*B10 + B20*B20 + B30*B30;
  const float M01 = B10*B11 + B20*B21 + B30*B31;
  const float M02 = B20*B22 + B30*B32;
  const float M03 = B30*B33;
  const float M11 = B11*B11 + B21*B21 + B31*B31;
  const float M12 = B21*B22 + B31*B32;
  const float M13 = B31*B33;
  const float M22 = B22*B22 + B32*B32;
  const float M23 = B32*B33;
  const float M33 = B33*B33;
  s[0]=M00;  s[1]=M01;  s[2]=M02;  s[3]=M03;
  s[4]=M01;  s[5]=M11;  s[6]=M12;  s[7]=M13;
  s[8]=M02;  s[9]=M12;  s[10]=M22; s[11]=M23;
  s[12]=M03; s[13]=M13; s[14]=M23; s[15]=M33;
}

__global__ __launch_bounds__(NTH)
void PlanLayer_41515153883715_kernel(const float* __restrict__ thb,
                                     const float* __restrict__ startb,
                                     const float* __restrict__ goalb,
                                     const float* __restrict__ sdfb,
                                     const float* __restrict__ qcinv,
                                     const float* __restrict__ obscov,
                                     const float* __restrict__ epsb,
                                     float* __restrict__ out)
{
  extern __shared__ float smem[];
  float* sD  = smem;                       // [NBAT][128][16] diag blocks -> S -> S^{-1}
  float* sU  = sD + NBAT*T1c*16;           // [NBAT][128][16] super-diag blocks U_t = -phi^T Qinv
  float* sZ  = sU + NBAT*T1c*16;           // [NBAT][128][4]  rhs -> z -> x
  float* sW  = sZ + NBAT*T1c*4;            // [NBAT][16]      transient W_t
  float* sEN = sW + NBAT*16;               // [NBAT][128]
  float* sEE = sEN + NBAT*T1c;             // [NBAT][128]

  const int tid = threadIdx.x;
  const int bq  = tid >> 7;                // batch-in-workgroup 0..3
  const int t   = tid & 127;               // time index 0..127
  const int b   = blockIdx.x * NBAT + bq;  // global batch

#define DLx(bb,tt,ii,jj) sD[((((bb)<<7)|(tt))<<4) + ((ii)<<2) + (jj)]
#define ULx(bb,tt,ii,jj) sU[((((bb)<<7)|(tt))<<4) + ((ii)<<2) + (jj)]
#define ZLx(bb,tt,ii)    sZ[((((bb)<<7)|(tt))<<2) + (ii)]
#define WLx(bb,kk,jj)    sW[((bb)<<4) + ((kk)<<2) + (jj)]

  // ----------------- Phase 1: build block-tridiagonal system -----------------
  {
    const float* th = thb + ((size_t)b*T1c + t)*SDc;
    __builtin_prefetch(th, 0, 0);                      // global_prefetch_b8
    __builtin_prefetch(sdfb + (size_t)b*HSc*WSc, 0, 0);
    const float th0=th[0], th1=th[1], th2=th[2], th3=th[3];

    float Dm[4][4];
    float c[4] = {0.f,0.f,0.f,0.f};
    float eN = 0.f, eE = 0.f;
    #pragma unroll
    for (int i=0;i<4;i++)
      #pragma unroll
      for (int j=0;j<4;j++) Dm[i][j] = (i==j) ? REGc : 0.f;

    if (t == 0) {          // prior factor
      const float* st = startb + (size_t)b*SDc;
      const float r0=th0-st[0], r1=th1-st[1], r2=th2-st[2], r3=th3-st[3];
      Dm[0][0]+=KSIc; Dm[1][1]+=KSIc; Dm[2][2]+=KSIc; Dm[3][3]+=KSIc;
      c[0]-=KSIc*r0; c[1]-=KSIc*r1; c[2]-=KSIc*r2; c[3]-=KSIc*r3;
      eN += KSIc*(r0*r0+r1*r1+r2*r2+r3*r3);
    }
    if (t == T1c-1) {      // goal factor
      const float* gl = goalb + (size_t)b*SDc;
      const float r0=th0-gl[0], r1=th1-gl[1], r2=th2-gl[2], r3=th3-gl[3];
      Dm[0][0]+=KSIc; Dm[1][1]+=KSIc; Dm[2][2]+=KSIc; Dm[3][3]+=KSIc;
      c[0]-=KSIc*r0; c[1]-=KSIc*r1; c[2]-=KSIc*r2; c[3]-=KSIc*r3;
      eN += KSIc*(r0*r0+r1*r1+r2*r2+r3*r3);
    }
    if (t < T1c-1) {       // GP factor g=t: contributes phi^T Q phi to D_t, -phi^T Q to U_t
      const float* tn = thb + ((size_t)b*T1c + t + 1)*SDc;
      float Q[4][4]; buildQ(qcinv + ((size_t)b*NGPc + t)*4, Q);
      float G[4][4]; // G = phi^T Q
      #pragma unroll
      for (int j=0;j<4;j++) {
        G[0][j] = Q[0][j];
        G[1][j] = Q[1][j];
        G[2][j] = DTc*Q[0][j] + Q[2][j];
        G[3][j] = DTc*Q[1][j] + Q[3][j];
      }
      const float r0=th0+DTc*th2-tn[0], r1=th1+DTc*th3-tn[1], r2=th2-tn[2], r3=th3-tn[3];
      #pragma unroll
      for (int i=0;i<4;i++) {
        Dm[i][0] += G[i][0];
        Dm[i][1] += G[i][1];
        Dm[i][2] += DTc*G[i][0] + G[i][2];   // P = G*phi
        Dm[i][3] += DTc*G[i][1] + G[i][3];
        c[i] -= G[i][0]*r0 + G[i][1]*r1 + G[i][2]*r2 + G[i][3]*r3;
        ULx(bq,t,i,0) = -G[i][0];
        ULx(bq,t,i,1) = -G[i][1];
        ULx(bq,t,i,2) = -G[i][2];
        ULx(bq,t,i,3) = -G[i][3];
      }
      const float s0 = Q[0][0]*r0+Q[0][1]*r1+Q[0][2]*r2+Q[0][3]*r3;
      const float s1 = Q[1][0]*r0+Q[1][1]*r1+Q[1][2]*r2+Q[1][3]*r3;
      const float s2 = Q[2][0]*r0+Q[2][1]*r1+Q[2][2]*r2+Q[2][3]*r3;
      const float s3 = Q[3][0]*r0+Q[3][1]*r1+Q[3][2]*r2+Q[3][3]*r3;
      eN += r0*s0 + r1*s1 + r2*s2 + r3*s3;
    }
    if (t >= 1) {          // GP factor g=t-1: contributes Q to D_t, +Q*r_gp[t-1] to rhs_t
      const float* tp = thb + ((size_t)b*T1c + t - 1)*SDc;
      float Qp[4][4]; buildQ(qcinv + ((size_t)b*NGPc + (t-1))*4, Qp);
      const float r0=tp[0]+DTc*tp[2]-th0, r1=tp[1]+DTc*tp[3]-th1, r2=tp[2]-th2, r3=tp[3]-th3;
      #pragma unroll
      for (int i=0;i<4;i++) {
        #pragma unroll
        for (int j=0;j<4;j++) Dm[i][j] += Qp[i][j];
        c[i] += Qp[i][0]*r0 + Qp[i][1]*r1 + Qp[i][2]*r2 + Qp[i][3]*r3;
      }
    }
    {                      // obstacle factor (bilinear SDF)
      const float* sdf = sdfb + (size_t)b*HSc*WSc;
      const float px = (th0 + 1.0f)*HLF;
      const float py = (th1 + 1.0f)*HLF;
      const float x0 = fminf(fmaxf(floorf(px), 0.f), (float)(WSc-2));
      const float y0 = fminf(fmaxf(floorf(py), 0.f), (float)(HSc-2));
      const int   xi = (int)x0, yi = (int)y0;
      const float tx = px - x0, ty = py - y0;
      const float v00 = sdf[yi*WSc + xi],     v01 = sdf[yi*WSc + xi + 1];
      const float v10 = sdf[(yi+1)*WSc + xi], v11 = sdf[(yi+1)*WSc + xi + 1];
      const float val = (1.f-ty)*((1.f-tx)*v00 + tx*v01) + ty*((1.f-tx)*v10 + tx*v11);
      const float gx  = ((1.f-ty)*(v01-v00) + ty*(v11-v10))*HLF;
      const float gy  = ((1.f-tx)*(v10-v00) + tx*(v11-v01))*HLF;
      const float eps = epsb[(size_t)b*T1c + t];
      const float ro  = fmaxf(eps - val, 0.f);
      const float act = (ro > 0.f) ? 1.f : 0.f;
      const float Jx = -act*gx, Jy = -act*gy;
      const float oc = obscov[(size_t)b*T1c + t];
      Dm[0][0] += oc*Jx*Jx; Dm[0][1] += oc*Jx*Jy;
      Dm[1][0] += oc*Jx*Jy; Dm[1][1] += oc*Jy*Jy;
      c[0] -= oc*ro*Jx; c[1] -= oc*ro*Jy;
      eN += oc*ro*ro;  eE += ro*ro;
    }
    #pragma unroll
    for (int i=0;i<4;i++) {
      #pragma unroll
      for (int j=0;j<4;j++) DLx(bq,t,i,j) = Dm[i][j];
      ZLx(bq,t,i) = c[i];
    }
    sEN[(bq<<7)+t] = eN;
    sEE[(bq<<7)+t] = eE;
  }
  __syncthreads();

  // ----------------- Phase 2: error reductions (deterministic tree) ----------
  for (int s2 = 64; s2 > 0; s2 >>= 1) {
    if (t < s2) {
      sEN[(bq<<7)+t] += sEN[(bq<<7)+t+s2];
      sEE[(bq<<7)+t] += sEE[(bq<<7)+t+s2];
    }
    __syncthreads();
  }

  // ----------------- Phase 3: block-tridiagonal Thomas (WMMA f32 16x16x4) ----
  // Forward: S_0=D_0; M_t=S_t^{-1}; W_t=M_t U_t; S_{t+1}=D_{t+1}-U_t^T W_t;
  //          z_{t+1}=c_{t+1}-W_t^T z_t.  4 batches packed per WMMA.
  for (int ts = 0; ts < T1c-1; ++ts) {
    if (tid < 4) inv4_spd(sD + ((((tid)<<7) | ts) << 4));   // M_ts in place
    __syncthreads();
    if (tid < 32) {  // wave0, EXEC all ones: W = M*U (block-diag harvest)
      const int nb = tid & 15, kb = (tid >> 4) << 1;
      const int cb = nb >> 2,  jj = nb & 3, half8 = (tid >> 4) << 3;
      v2f a, bv; v8f acc = {0.f,0.f,0.f,0.f,0.f,0.f,0.f,0.f};
      a[0]  = DLx(cb, ts, jj, kb);  a[1]  = DLx(cb, ts, jj, kb+1);   // A[m][k]=M[i][k]
      bv[0] = ULx(cb, ts, kb, jj);  bv[1] = ULx(cb, ts, kb+1, jj);   // B[k][n]=U[k][j]
      acc = __builtin_amdgcn_wmma_f32_16x16x4_f32(false, a, false, bv,
                                                  (short)0, acc, false, false);
      #pragma unroll
      for (int r = 0; r < 8; ++r) {
        const int m = r + half8;
        if ((m >> 2) == cb) WLx(cb, m & 3, jj) = acc[r];
      }
    }
    __syncthreads();
    if (tid < 32) {  // wave0: S_{t+1} -= U^T * W (block-diag harvest)
      const int nb = tid & 15, kb = (tid >> 4) << 1;
      const int cb = nb >> 2,  jj = nb & 3, half8 = (tid >> 4) << 3;
      v2f a, bv; v8f acc = {0.f,0.f,0.f,0.f,0.f,0.f,0.f,0.f};
      a[0]  = ULx(cb, ts, kb, jj);  a[1]  = ULx(cb, ts, kb+1, jj);   // A[m][k]=U[k][i]
      bv[0] = WLx(cb, kb, jj);      bv[1] = WLx(cb, kb+1, jj);       // B[k][n]=W[k][j]
      acc = __builtin_amdgcn_wmma_f32_16x16x4_f32(false, a, false, bv,
                                                  (short)0, acc, false, false);
      #pragma unroll
      for (int r = 0; r < 8; ++r) {
        const int m = r + half8;
        if ((m >> 2) == cb) DLx(cb, ts+1, m & 3, jj) -= acc[r];
      }
    } else if (tid < 48) {  // wave1: z_{t+1} -= W^T z_t
      const int u = tid - 32, zb = u >> 2, ii = u & 3;
      const float s = WLx(zb,0,ii)*ZLx(zb,ts,0) + WLx(zb,1,ii)*ZLx(zb,ts,1)
                    + WLx(zb,2,ii)*ZLx(zb,ts,2) + WLx(zb,3,ii)*ZLx(zb,ts,3);
      ZLx(zb, ts+1, ii) -= s;
    }
    __syncthreads();
  }

  if (tid < 4) inv4_spd(sD + ((((tid)<<7) | (T1c-1)) << 4));
  __syncthreads();
  if (tid < 4) {   // x_{127} = M_{127} z_{127}
    const int bb = tid;
    const float z0=ZLx(bb,T1c-1,0), z1=ZLx(bb,T1c-1,1), z2=ZLx(bb,T1c-1,2), z3=ZLx(bb,T1c-1,3);
    const float x0=DLx(bb,T1c-1,0,0)*z0+DLx(bb,T1c-1,0,1)*z1+DLx(bb,T1c-1,0,2)*z2+DLx(bb,T1c-1,0,3)*z3;
    const float x1=DLx(bb,T1c-1,1,0)*z0+DLx(bb,T1c-1,1,1)*z1+DLx(bb,T1c-1,1,2)*z2+DLx(bb,T1c-1,1,3)*z3;
    const float x2=DLx(bb,T1c-1,2,0)*z0+DLx(bb,T1c-1,2,1)*z1+DLx(bb,T1c-1,2,2)*z2+DLx(bb,T1c-1,2,3)*z3;
    const float x3=DLx(bb,T1c-1,3,0)*z0+DLx(bb,T1c-1,3,1)*z1+DLx(bb,T1c-1,3,2)*z2+DLx(bb,T1c-1,3,3)*z3;
    ZLx(bb,T1c-1,0)=x0; ZLx(bb,T1c-1,1)=x1; ZLx(bb,T1c-1,2)=x2; ZLx(bb,T1c-1,3)=x3;
  }
  __syncthreads();

  // Backward: x_t = M_t (z_t - U_t x_{t+1})
  for (int ts = T1c-2; ts >= 0; --ts) {
    if (tid < 4) {
      const int bb = tid;
      const float xn0=ZLx(bb,ts+1,0), xn1=ZLx(bb,ts+1,1), xn2=ZLx(bb,ts+1,2), xn3=ZLx(bb,ts+1,3);
      float tm[4];
      #pragma unroll
      for (int i=0;i<4;i++)
        tm[i] = ZLx(bb,ts,i) - (ULx(bb,ts,i,0)*xn0 + ULx(bb,ts,i,1)*xn1
                              + ULx(bb,ts,i,2)*xn2 + ULx(bb,ts,i,3)*xn3);
      #pragma unroll
      for (int i=0;i<4;i++)
        ZLx(bb,ts,i) = DLx(bb,ts,i,0)*tm[0] + DLx(bb,ts,i,1)*tm[1]
                     + DLx(bb,ts,i,2)*tm[2] + DLx(bb,ts,i,3)*tm[3];
    }
    __syncthreads();
  }

  // ----------------- Phase 4: write outputs ----------------------------------
  {
    float* o = out + ((size_t)b*T1c + t)*SDc;
    o[0]=ZLx(bq,t,0); o[1]=ZLx(bq,t,1); o[2]=ZLx(bq,t,2); o[3]=ZLx(bq,t,3);
    if (t == 0) {
      out[(size_t)Bsz*T1c*SDc + b]       = sEN[bq<<7];
      out[(size_t)Bsz*T1c*SDc + Bsz + b] = sEE[bq<<7];
    }
  }
}

extern "C" void kernel_launch(void* const* d_in, const int* in_sizes, int n_in,
                              void* d_out, int out_size, void* d_ws, size_t ws_size,
                              hipStream_t stream) {
  (void)in_sizes; (void)n_in; (void)d_ws; (void)ws_size; (void)out_size;
  const float* thb    = (const float*)d_in[0];
  const float* startb = (const float*)d_in[1];
  const float* goalb  = (const float*)d_in[2];
  // d_in[3] = imb (unused by the reference forward pass)
  const float* sdfb   = (const float*)d_in[4];
  const float* qcinv  = (const float*)d_in[5];
  const float* obscov = (const float*)d_in[6];
  const float* epsb   = (const float*)d_in[7];
  float* out = (float*)d_out;

  const size_t smem_bytes =
      (size_t)(NBAT*T1c*16 /*sD*/ + NBAT*T1c*16 /*sU*/ + NBAT*T1c*4 /*sZ*/
               + NBAT*16 /*sW*/ + NBAT*T1c /*sEN*/ + NBAT*T1c /*sEE*/) * sizeof(float);

  PlanLayer_41515153883715_kernel<<<dim3(Bsz/NBAT), dim3(NTH), smem_bytes, stream>>>(
      thb, startb, goalb, sdfb, qcinv, obscov, epsb, out);
}